// differentiableSuperpixelEmbedding_60473139527755
// MI455X (gfx1250) — compile-verified
//
#include <hip/hip_runtime.h>

// ---------------- problem constants (from reference) ----------------
#define NBATCH   32
#define NCH      3
#define NPIX     147456      // 384*384
#define NSEG     256
#define NPPS     400         // MAX_PIXELS per segment
#define KDIM     1200        // NCH*NPPS
#define EDIM     768
#define MDIM     (NBATCH*NSEG)   // 8192 rows of feats
#define NTILES   144
#define TSZ      1024        // NTILES*TSZ == NPIX

// ws layout: feats[MDIM*KDIM] floats, then hist[NBATCH*NTILES*NSEG] ints
#define FEATS_BYTES ((size_t)MDIM * KDIM * sizeof(float))   // 39,321,600

typedef float v2f __attribute__((ext_vector_type(2)));
typedef float v8f __attribute__((ext_vector_type(8)));

// ---------------- kernel 0: zero the feats scratch ----------------
__global__ void k_zero_feats(float4* feats4, int n4) {
    int i = blockIdx.x * blockDim.x + threadIdx.x;
    int stride = gridDim.x * blockDim.x;
    float4 z = make_float4(0.f, 0.f, 0.f, 0.f);
    for (; i < n4; i += stride) feats4[i] = z;
}

// ---------------- kernel 1: per-(batch,tile) segment histogram ----------------
__global__ void k_hist(const int* __restrict__ segments, int* __restrict__ hist) {
    __shared__ int lh[NSEG];
    const int b = blockIdx.x / NTILES;
    const int t = blockIdx.x % NTILES;
    const int tid = threadIdx.x;
    lh[tid] = 0;
    __syncthreads();
    const int base = b * NPIX + t * TSZ;
#pragma unroll
    for (int i = 0; i < 4; ++i) {
        int s = segments[base + tid + i * 256];
        atomicAdd(&lh[s], 1);
    }
    __syncthreads();
    hist[(b * NTILES + t) * NSEG + tid] = lh[tid];
}

// ---------------- kernel 2: in-place exclusive prefix over tiles ----------------
__global__ void k_prefix(int* __restrict__ hist) {
    const int b = blockIdx.x;
    const int s = threadIdx.x;
    int running = 0;
    for (int t0 = 0; t0 < NTILES; t0 += 16) {
        int h[16];
#pragma unroll
        for (int i = 0; i < 16; ++i)
            h[i] = hist[(b * NTILES + (t0 + i)) * NSEG + s];
#pragma unroll
        for (int i = 0; i < 16; ++i) {
            hist[(b * NTILES + (t0 + i)) * NSEG + s] = running;
            running += h[i];
        }
    }
}

// ---------------- kernel 3: scan-order local rank + scatter ----------------
__global__ void k_scatter(const float* __restrict__ img,
                          const int* __restrict__ segments,
                          const int* __restrict__ base,   // exclusive prefix
                          float* __restrict__ feats) {
    __shared__ int segl[TSZ];
    __shared__ int rnk[TSZ];
    const int b = blockIdx.x / NTILES;
    const int t = blockIdx.x % NTILES;
    const int tid = threadIdx.x;
    const int poff = b * NPIX + t * TSZ;
#pragma unroll
    for (int i = 0; i < 4; ++i)
        segl[tid + i * 256] = segments[poff + tid + i * 256];
    __syncthreads();
    // owner loop: exactly one thread claims each pixel -> scan-order rank
    int cnt = 0;
    for (int p = 0; p < TSZ; ++p) {
        if (segl[p] == tid) { rnk[p] = cnt; ++cnt; }
    }
    __syncthreads();
    const int* tile_base = base + (b * NTILES + t) * NSEG;
#pragma unroll
    for (int i = 0; i < 4; ++i) {
        const int p = tid + i * 256;
        const int s = segl[p];
        const int r = tile_base[s] + rnk[p];
        if (r < NPPS) {
            const int n = t * TSZ + p;
#pragma unroll
            for (int c = 0; c < NCH; ++c)
                feats[(((size_t)b * NSEG + s) * NCH + c) * NPPS + r] =
                    img[((size_t)(b * NCH + c)) * NPIX + n];
        }
    }
}

// ---------------- kernel 4: WMMA f32 GEMM ----------------
// out[M=8192, E=768] = feats[M, K=1200] * W[E, K]^T + bias[E]
// Block tile 128(M) x 128(N), KC=48. 8 waves arranged 4(M) x 2(N); each wave
// computes 32x64 via 2x4 accumulators (8 x v8f). Register double-buffered
// global->LDS staging to overlap prefetch with the 96-WMMA compute body.
#define BM 128
#define BN 128
#define KC 48
#define LDP 52   // padded LDS row stride in floats (208B, 16B aligned)

__device__ __forceinline__ void gemm_compute(const float* __restrict__ ldsA,
                                             const float* __restrict__ ldsB,
                                             int wm, int wn, int row16, int half,
                                             v8f acc[2][4]) {
#pragma unroll
    for (int kk = 0; kk < KC; kk += 4) {
        v2f a[2];
#pragma unroll
        for (int mi = 0; mi < 2; ++mi) {
            const float* ap = &ldsA[(wm + mi * 16 + row16) * LDP + kk + half * 2];
            a[mi].x = ap[0]; a[mi].y = ap[1];
        }
#pragma unroll
        for (int ni = 0; ni < 4; ++ni) {
            v2f bf;
            const float* bp = &ldsB[(wn + ni * 16 + row16) * LDP + kk + half * 2];
            bf.x = bp[0]; bf.y = bp[1];
#pragma unroll
            for (int mi = 0; mi < 2; ++mi)
                acc[mi][ni] = __builtin_amdgcn_wmma_f32_16x16x4_f32(
                    false, a[mi], false, bf, (short)0, acc[mi][ni], false, false);
        }
    }
}

__global__ void __launch_bounds__(256)
k_gemm(const float* __restrict__ feats, const float* __restrict__ W,
       const float* __restrict__ bias, float* __restrict__ out) {
    __shared__ float ldsA[BM * LDP];   // [row 0..127][k 0..47]
    __shared__ float ldsB[BN * LDP];   // [col 0..127][k 0..47]

    const int tid  = threadIdx.x;
    const int lane = tid & 31;
    const int wave = tid >> 5;
    const int row16 = lane & 15;
    const int half  = lane >> 4;        // per-ISA lane-half for A(K)/C(M) split
    const int wm = (wave >> 1) * 32;    // wave's M base within block (4 waves in M)
    const int wn = (wave & 1) * 64;     // wave's N base within block (2 waves in N)
    const int m0 = blockIdx.y * BM;
    const int n0 = blockIdx.x * BN;

    v8f acc[2][4];
#pragma unroll
    for (int mi = 0; mi < 2; ++mi)
#pragma unroll
        for (int ni = 0; ni < 4; ++ni)
            acc[mi][ni] = (v8f){0.f,0.f,0.f,0.f,0.f,0.f,0.f,0.f};

    float4 ra[6], rb[6];
    // prologue: fetch chunk 0 (A: 128 rows x 12 float4, B: 128 cols x 12 float4)
#pragma unroll
    for (int i = 0; i < 6; ++i) {
        int id = tid + i * 256, r = id / 12, cv = id % 12;
        ra[i] = *(const float4*)&feats[(size_t)(m0 + r) * KDIM + cv * 4];
        rb[i] = *(const float4*)&W[(size_t)(n0 + r) * KDIM + cv * 4];
    }
#pragma unroll
    for (int i = 0; i < 6; ++i) {
        int id = tid + i * 256, r = id / 12, cv = id % 12;
        *(float4*)&ldsA[r * LDP + cv * 4] = ra[i];
        *(float4*)&ldsB[r * LDP + cv * 4] = rb[i];
    }
    __syncthreads();

    for (int k0 = KC; k0 < KDIM; k0 += KC) {
        // prefetch next chunk into registers (overlaps with WMMA compute below)
#pragma unroll
        for (int i = 0; i < 6; ++i) {
            int id = tid + i * 256, r = id / 12, cv = id % 12;
            ra[i] = *(const float4*)&feats[(size_t)(m0 + r) * KDIM + k0 + cv * 4];
            rb[i] = *(const float4*)&W[(size_t)(n0 + r) * KDIM + k0 + cv * 4];
        }
        gemm_compute(ldsA, ldsB, wm, wn, row16, half, acc);
        __syncthreads();   // everyone done reading current LDS chunk
#pragma unroll
        for (int i = 0; i < 6; ++i) {
            int id = tid + i * 256, r = id / 12, cv = id % 12;
            *(float4*)&ldsA[r * LDP + cv * 4] = ra[i];
            *(float4*)&ldsB[r * LDP + cv * 4] = rb[i];
        }
        __syncthreads();   // new chunk visible
    }
    gemm_compute(ldsA, ldsB, wm, wn, row16, half, acc);

    // epilogue: C layout — VGPR r: M = r + half*8, N = row16
#pragma unroll
    for (int ni = 0; ni < 4; ++ni) {
        const int gcol = n0 + wn + ni * 16 + row16;
        const float bv = bias[gcol];
#pragma unroll
        for (int mi = 0; mi < 2; ++mi) {
#pragma unroll
            for (int r = 0; r < 8; ++r) {
                const int grow = m0 + wm + mi * 16 + half * 8 + r;
                out[(size_t)grow * EDIM + gcol] = acc[mi][ni][r] + bv;
            }
        }
    }
}

// ---------------- launcher ----------------
extern "C" void kernel_launch(void* const* d_in, const int* in_sizes, int n_in,
                              void* d_out, int out_size, void* d_ws, size_t ws_size,
                              hipStream_t stream) {
    const float* img      = (const float*)d_in[0];   // [32,3,384,384]
    const int*   segments = (const int*)  d_in[1];   // [32,384,384]
    const float* proj_w   = (const float*)d_in[2];   // [768,1200]
    const float* proj_b   = (const float*)d_in[3];   // [768]
    float* out = (float*)d_out;                      // [32,256,768]

    float* feats = (float*)d_ws;
    int*   hist  = (int*)((char*)d_ws + FEATS_BYTES);

    // 0) zero feats
    {
        int n4 = (MDIM * KDIM) / 4;   // 2,457,600 float4
        k_zero_feats<<<2400, 256, 0, stream>>>((float4*)feats, n4);
    }
    // 1) histograms
    k_hist<<<NBATCH * NTILES, 256, 0, stream>>>(segments, hist);
    // 2) exclusive prefix over tiles
    k_prefix<<<NBATCH, 256, 0, stream>>>(hist);
    // 3) rank + scatter into feats
    k_scatter<<<NBATCH * NTILES, 256, 0, stream>>>(img, segments, hist, feats);
    // 4) WMMA GEMM + bias
    {
        dim3 grid(EDIM / BN, MDIM / BM);   // (6, 64)
        k_gemm<<<grid, 256, 0, stream>>>(feats, proj_w, proj_b, out);
    }
}